// GenericAntisymmetry_5677946766059
// MI455X (gfx1250) — compile-verified
//
#include <hip/hip_runtime.h>
#include <hip/hip_bf16.h>
#include <math.h>

// ---------------------------------------------------------------------------
// Problem constants (fixed by the reference)
// ---------------------------------------------------------------------------
#define B_      32
#define NELEC_  10
#define NSPIN_  5
#define D_      3
#define NION_   4
#define DBF_    16
#define P_      120          // 5!
#define H_      64
#define KDIM_   160          // NELEC_*DBF_
#define NTILES_PER_B_ 900    // 14400 / 16
#define NTILES_ (B_ * NTILES_PER_B_)   // 28800

// workspace layout (in floats)
#define WS_STREAM_OFF 0                 // 32*10*16 = 5120
#define WS_WR0F_OFF   5120              // 40*4*32*2 = 10240 (pre-swizzled Wr0 frags)
#define WS_WR1F_OFF   15360             // 16*4*32*2 = 4096  (pre-swizzled Wr1 frags)
#define WS_PSI_OFF    19456             // 32 accumulators

typedef __attribute__((ext_vector_type(2))) float v2f;
typedef __attribute__((ext_vector_type(8))) float v8f;

// ---------------------------------------------------------------------------
// CDNA5 hardware tanh (V_TANH_F32, single TRANS op; co-executes with WMMA).
// Falls back to inline asm if the builtin is not declared by this toolchain.
// ---------------------------------------------------------------------------
__device__ __forceinline__ float hw_tanh(float x) {
#if __has_builtin(__builtin_amdgcn_tanhf)
    return __builtin_amdgcn_tanhf(x);
#else
    float r;
    asm("v_tanh_f32 %0, %1" : "=v"(r) : "v"(x));
    return r;
#endif
}

// ---------------------------------------------------------------------------
// Lexicographic permutation decode for n=5 (matches itertools.permutations).
// Parity of permutation == sum of factorial digits.
// ---------------------------------------------------------------------------
__device__ __forceinline__ void perm5_decode(int idx, int perm[5]) {
    int elems[5] = {0, 1, 2, 3, 4};
    int d0 = idx / 24; idx -= d0 * 24;
    int d1 = idx / 6;  idx -= d1 * 6;
    int d2 = idx / 2;  idx -= d2 * 2;
    int dg[5] = {d0, d1, d2, idx, 0};
#pragma unroll
    for (int i = 0; i < 5; ++i) {
        int d = dg[i];
        perm[i] = elems[d];
#pragma unroll
        for (int j = 0; j < 4; ++j)
            if (j >= d) elems[j] = elems[j + 1];
    }
}

__device__ __forceinline__ float perm5_sign(int idx) {
    int d0 = idx / 24; idx -= d0 * 24;
    int d1 = idx / 6;  idx -= d1 * 6;
    int d2 = idx / 2;  int d3 = idx - d2 * 2;
    return ((d0 + d1 + d2 + d3) & 1) ? -1.0f : 1.0f;
}

// ---------------------------------------------------------------------------
// Kernel 1: pre-swizzle Wr0/Wr1 into WMMA B-fragment order + zero psi acc.
// B-fragment layout for V_WMMA_F32_16X16X4_F32 (4xK, 16xN):
//   lane = {hi=lane>>4, lo=lane&15}, vgpr j in {0,1}:  element K=2*hi+j, N=lo
// ---------------------------------------------------------------------------
__global__ void prep_kernel(const float* __restrict__ Wr0,
                            const float* __restrict__ Wr1,
                            float* __restrict__ ws) {
    int t = blockIdx.x * blockDim.x + threadIdx.x;
    if (t < 10240) {  // Wr0 frags: 40 ksteps * 4 ntiles * 32 lanes * 2
        int j = t & 1, lane = (t >> 1) & 31, n = (t >> 6) & 3, ki = t >> 8;
        int hi = lane >> 4, lo = lane & 15;
        int k = 4 * ki + 2 * hi + j;
        int col = n * 16 + lo;
        ws[WS_WR0F_OFF + t] = Wr0[k * H_ + col];
    }
    if (t < 4096) {   // Wr1 frags: 16 ksteps * 4 ntiles * 32 lanes * 2
        int j = t & 1, lane = (t >> 1) & 31, n = (t >> 6) & 3, kj = t >> 8;
        int hi = lane >> 4, lo = lane & 15;
        int k = 4 * kj + 2 * hi + j;
        int col = n * 16 + lo;
        ws[WS_WR1F_OFF + t] = Wr1[k * H_ + col];
    }
    if (t < B_) ws[WS_PSI_OFF + t] = 0.0f;
}

// ---------------------------------------------------------------------------
// Kernel 2: per-electron feature streams (tiny: 320 rows, two 16x16 tanh MLPs)
// Uses libm tanhf for maximum accuracy at the source of the pipeline.
// ---------------------------------------------------------------------------
__global__ void stream_kernel(const float* __restrict__ elec,
                              const float* __restrict__ ion,
                              const float* __restrict__ W1,
                              const float* __restrict__ b1,
                              const float* __restrict__ W2,
                              const float* __restrict__ b2,
                              float* __restrict__ ws) {
    int t = blockIdx.x * blockDim.x + threadIdx.x;
    if (t >= B_ * NELEC_) return;
    int b = t / NELEC_, e = t - b * NELEC_;
    float ex = elec[t * 3 + 0], ey = elec[t * 3 + 1], ez = elec[t * 3 + 2];
    float in[16];
#pragma unroll
    for (int i = 0; i < NION_; ++i) {
        float dx = ex - ion[i * 3 + 0];
        float dy = ey - ion[i * 3 + 1];
        float dz = ez - ion[i * 3 + 2];
        float d = sqrtf(dx * dx + dy * dy + dz * dz);
        in[i * 4 + 0] = dx; in[i * 4 + 1] = dy;
        in[i * 4 + 2] = dz; in[i * 4 + 3] = d;
    }
    float h[16];
#pragma unroll
    for (int jj = 0; jj < 16; ++jj) {
        float s = b1[jj];
#pragma unroll
        for (int k = 0; k < 16; ++k) s += in[k] * W1[k * 16 + jj];
        h[jj] = tanhf(s);
    }
#pragma unroll
    for (int jj = 0; jj < 16; ++jj) {
        float s = b2[jj];
#pragma unroll
        for (int k = 0; k < 16; ++k) s += h[k] * W2[k * 16 + jj];
        ws[WS_STREAM_OFF + b * KDIM_ + e * DBF_ + jj] = h[jj] + tanhf(s);
    }
}

// ---------------------------------------------------------------------------
// Kernel 3: main WMMA kernel. One wave per 16-row M tile of the virtual
// (B*P*P, 160) matrix; rows gathered on-the-fly from LDS stream table.
// Layer1: (16x160)@(160x64) via 40 k-steps * 4 n-tiles of wmma_f32_16x16x4_f32
// Layer2: residual (16x64)@(64x64)  via 16 k-steps * 4 n-tiles
// Head:   64->1 dot + sign-weighted cross-lane reduction -> atomicAdd psi[b]
// Nonlinearities use hardware V_TANH_F32 (TRANS pipe, overlaps with WMMA).
// ---------------------------------------------------------------------------
__global__ void __launch_bounds__(256)
antisym_kernel(const float* __restrict__ ws,
               const float* __restrict__ br0,
               const float* __restrict__ br1,
               const float* __restrict__ Wout,
               const float* __restrict__ bout,
               float* __restrict__ psiAcc) {
    __shared__ float lds_stream[B_ * KDIM_];   // 5120 floats = 20 KB
    __shared__ float lds_hr[8][16 * H_];       // per-wave staging, 32 KB

    const float* wr0f = ws + WS_WR0F_OFF;
    const float* wr1f = ws + WS_WR1F_OFF;

    int tid = threadIdx.x;
    for (int i = tid; i < B_ * KDIM_; i += 256) lds_stream[i] = ws[WS_STREAM_OFF + i];
    __syncthreads();

    const int wave = tid >> 5;
    const int lane = tid & 31;
    const int lo = lane & 15;
    const int hi = lane >> 4;
    const int koff = 2 * hi;     // K sub-offset held by this half of the wave

    float br0l[4], br1l[4], woutl[4];
#pragma unroll
    for (int n = 0; n < 4; ++n) {
        br0l[n]  = br0[n * 16 + lo];
        br1l[n]  = br1[n * 16 + lo];
        woutl[n] = Wout[n * 16 + lo];
    }
    const float bout0 = bout[0];
    float* hrl = lds_hr[wave];

    const v8f zero8 = {0.f, 0.f, 0.f, 0.f, 0.f, 0.f, 0.f, 0.f};

    int gw = blockIdx.x * 8 + wave;
    int nw = gridDim.x * 8;

    for (int t = gw; t < NTILES_; t += nw) {
        int b  = t / NTILES_PER_B_;
        int tb = t - b * NTILES_PER_B_;

        // A-row owned by this lane pair: M = lo ; decode its (p,q) slot map
        int idx = tb * 16 + lo;
        int p = idx / P_, q = idx - (idx / P_) * P_;
        int permp[5], permq[5];
        perm5_decode(p, permp);
        perm5_decode(q, permq);
        int sm[10];
#pragma unroll
        for (int s = 0; s < 5; ++s) { sm[s] = permp[s]; sm[5 + s] = NSPIN_ + permq[s]; }

        // sign for the C/D-layout row this lane will accumulate (M = lo + 8*hi,
        // only lanes with lo < 8 accumulate)
        int idx2 = tb * 16 + (lo & 7) + 8 * hi;
        int p2 = idx2 / P_, q2 = idx2 - (idx2 / P_) * P_;
        float mysign = (lo < 8) ? perm5_sign(p2) * perm5_sign(q2) : 0.0f;

        const float* sb = &lds_stream[b * KDIM_];

        // ---------------- layer 1: (16x160) @ (160x64) ----------------
        v8f acc[4] = {zero8, zero8, zero8, zero8};
#pragma unroll 4
        for (int ki = 0; ki < 40; ++ki) {
            int blk = sm[ki >> 2];                       // electron slot for this k-step
            int off = blk * DBF_ + (ki & 3) * 4 + koff;  // 8B-aligned
            v2f a = *(const v2f*)&sb[off];
#pragma unroll
            for (int n = 0; n < 4; ++n) {
                v2f bf = *(const v2f*)&wr0f[((ki * 4 + n) * 32 + lane) * 2];
                acc[n] = __builtin_amdgcn_wmma_f32_16x16x4_f32(
                    false, a, false, bf, (short)0, acc[n], false, false);
            }
        }

        // bias + hw tanh, keep in regs (residual) and stage to LDS for layer-2 A
        v8f hr1[4];
#pragma unroll
        for (int n = 0; n < 4; ++n) {
#pragma unroll
            for (int r = 0; r < 8; ++r) {
                float v = hw_tanh(acc[n][r] + br0l[n]);
                hr1[n][r] = v;
                hrl[(r + 8 * hi) * H_ + n * 16 + lo] = v;   // row-major 16x64
            }
        }
        asm volatile("" ::: "memory");   // keep DS store->load program order

        // ---------------- layer 2: (16x64) @ (64x64) ----------------
        v8f acc2[4] = {zero8, zero8, zero8, zero8};
#pragma unroll
        for (int kj = 0; kj < 16; ++kj) {
            v2f a = *(const v2f*)&hrl[lo * H_ + kj * 4 + koff];
#pragma unroll
            for (int n = 0; n < 4; ++n) {
                v2f bf = *(const v2f*)&wr1f[((kj * 4 + n) * 32 + lane) * 2];
                acc2[n] = __builtin_amdgcn_wmma_f32_16x16x4_f32(
                    false, a, false, bf, (short)0, acc2[n], false, false);
            }
        }

        // ------------- head: residual + 64->1 dot + signed reduce -------------
        float psil = 0.0f;
#pragma unroll
        for (int r = 0; r < 8; ++r) {
            float part = 0.0f;
#pragma unroll
            for (int n = 0; n < 4; ++n)
                part += (hr1[n][r] + hw_tanh(acc2[n][r] + br1l[n])) * woutl[n];
            part += __shfl_xor(part, 1, 32);
            part += __shfl_xor(part, 2, 32);
            part += __shfl_xor(part, 4, 32);
            part += __shfl_xor(part, 8, 32);
            if (lo == r) psil += mysign * (part + bout0);  // row M = r + 8*hi
        }
        psil += __shfl_xor(psil, 16, 32);
        psil += __shfl_xor(psil, 8, 32);
        psil += __shfl_xor(psil, 4, 32);
        psil += __shfl_xor(psil, 2, 32);
        psil += __shfl_xor(psil, 1, 32);
        if (lane == 0) atomicAdd(&psiAcc[b], psil);
        asm volatile("" ::: "memory");   // hrl reused next iteration
    }
}

// ---------------------------------------------------------------------------
// Kernel 4: sign/log readout + Jastrow (32 threads)
// ---------------------------------------------------------------------------
__global__ void finalize_kernel(const float* __restrict__ elec,
                                const float* __restrict__ alpha,
                                const float* __restrict__ ws,
                                float* __restrict__ out) {
    int b = threadIdx.x;
    if (b >= B_) return;
    float psi = ws[WS_PSI_OFF + b];
    float js = 0.0f;
#pragma unroll
    for (int i = 0; i < NELEC_; ++i) {
        float xi = elec[(b * NELEC_ + i) * 3 + 0];
        float yi = elec[(b * NELEC_ + i) * 3 + 1];
        float zi = elec[(b * NELEC_ + i) * 3 + 2];
#pragma unroll
        for (int j = i + 1; j < NELEC_; ++j) {
            float dx = xi - elec[(b * NELEC_ + j) * 3 + 0];
            float dy = yi - elec[(b * NELEC_ + j) * 3 + 1];
            float dz = zi - elec[(b * NELEC_ + j) * 3 + 2];
            js += sqrtf(dx * dx + dy * dy + dz * dz);
        }
    }
    float sg = (psi > 0.0f) ? 1.0f : ((psi < 0.0f) ? -1.0f : 0.0f);
    out[b]       = sg;
    out[B_ + b]  = logf(fabsf(psi) + 1e-30f) - alpha[0] * js;
}

// ---------------------------------------------------------------------------
extern "C" void kernel_launch(void* const* d_in, const int* in_sizes, int n_in,
                              void* d_out, int out_size, void* d_ws, size_t ws_size,
                              hipStream_t stream) {
    const float* elec  = (const float*)d_in[0];
    const float* ion   = (const float*)d_in[1];
    const float* W1    = (const float*)d_in[2];
    const float* b1    = (const float*)d_in[3];
    const float* W2    = (const float*)d_in[4];
    const float* b2    = (const float*)d_in[5];
    const float* Wr0   = (const float*)d_in[6];
    const float* br0   = (const float*)d_in[7];
    const float* Wr1   = (const float*)d_in[8];
    const float* br1   = (const float*)d_in[9];
    const float* Wout  = (const float*)d_in[10];
    const float* bout  = (const float*)d_in[11];
    const float* alpha = (const float*)d_in[12];
    float* out = (float*)d_out;
    float* ws  = (float*)d_ws;

    prep_kernel<<<40, 256, 0, stream>>>(Wr0, Wr1, ws);
    stream_kernel<<<10, 32, 0, stream>>>(elec, ion, W1, b1, W2, b2, ws);
    antisym_kernel<<<120, 256, 0, stream>>>(ws, br0, br1, Wout, bout,
                                            ws + WS_PSI_OFF);
    finalize_kernel<<<1, 32, 0, stream>>>(elec, alpha, ws, out);
}